// MaskGenerate_51685636440888
// MI455X (gfx1250) — compile-verified
//
#include <hip/hip_runtime.h>
#include <stdint.h>

// Stratified mask generation for MI455X (gfx1250).
// One 256-thread workgroup per (b,t) row of N=1024 scores.
// CDNA5 paths: global_load_async_to_lds_b128 + global_store_async_from_lds_b128
// (ASYNCcnt) for fully-coalesced HBM traffic, LDS-resident bitonic sorts
// (ds ops), wave32 execution.

#define NROW 1024
#define SEG 256
#define THREADS 256

__device__ __forceinline__ unsigned hash32(unsigned x) {
  x ^= x >> 16; x *= 0x7feb352dU;
  x ^= x >> 15; x *= 0x846ca68bU;
  x ^= x >> 16; return x;
}

__global__ __launch_bounds__(THREADS)
void stratified_mask_kernel(const float* __restrict__ scores,
                            const float* __restrict__ p_mask_ratio,
                            float* __restrict__ out) {
  __shared__ float              s_scores[NROW];   // 4 KB (reused as out row)
  __shared__ unsigned long long s_key[NROW];      // 8 KB
  __shared__ unsigned short     s_order[NROW];    // 2 KB

  const unsigned tid = threadIdx.x;
  const unsigned row = blockIdx.x;                // b*T + t, 0..8191
  const float* rowp = scores + (size_t)row * NROW;

  // ---------------------------------------------------------------
  // Phase 0: async global -> LDS copy of the score row (ASYNCcnt).
  // Each lane moves 16 bytes; 256 lanes cover the whole 4 KB row.
  // ---------------------------------------------------------------
  {
    unsigned ldsOff  = (unsigned)(uintptr_t)(&s_scores[0]) + tid * 16u;
    unsigned byteOff = tid * 16u;
    unsigned long long base = (unsigned long long)(uintptr_t)rowp;
    asm volatile("global_load_async_to_lds_b128 %0, %1, %2"
                 :: "v"(ldsOff), "v"(byteOff), "s"(base)
                 : "memory");
    asm volatile("s_wait_asynccnt 0" ::: "memory");
  }
  __syncthreads();

  // ---------------------------------------------------------------
  // Phase 1: 64-bit keys for stable DESCENDING score sort.
  // orderable(f) is monotone-ascending over float order; invert it so
  // an ascending u64 sort yields descending scores, ties by index asc.
  // ---------------------------------------------------------------
#pragma unroll
  for (int q = 0; q < 4; ++q) {
    unsigned j = tid + q * THREADS;
    unsigned bits = __float_as_uint(s_scores[j]);
    unsigned ordv = bits ^ ((bits & 0x80000000u) ? 0xFFFFFFFFu : 0x80000000u);
    s_key[j] = (((unsigned long long)(~ordv)) << 32) | (unsigned long long)j;
  }

  // Bitonic sort of 1024 u64 keys; 512 compare-exchanges per step,
  // 55 steps, fully unrolled (bounds are compile-time constants).
#pragma unroll
  for (unsigned size = 2; size <= NROW; size <<= 1) {
#pragma unroll
    for (unsigned stride = size >> 1; stride > 0; stride >>= 1) {
      __syncthreads();
#pragma unroll
      for (int q = 0; q < 2; ++q) {
        unsigned p  = tid + q * THREADS;                       // 0..511
        unsigned i  = ((p & ~(stride - 1u)) << 1) | (p & (stride - 1u));
        unsigned jj = i + stride;
        bool up = ((i & size) == 0u);
        unsigned long long a = s_key[i], b = s_key[jj];
        bool sw = up ? (a > b) : (b > a);
        if (sw) { s_key[i] = b; s_key[jj] = a; }
      }
    }
  }
  __syncthreads();

  // order[j] = original index of the element at sorted position j.
#pragma unroll
  for (int q = 0; q < 4; ++q) {
    unsigned j = tid + q * THREADS;
    s_order[j] = (unsigned short)(s_key[j] & 1023u);
  }
  __syncthreads();

  // ---------------------------------------------------------------
  // Phase 2: per-sorted-position randoms; stable ascending rank inside
  // each 256-element segment via four parallel bitonic sorts.
  // key = (rand32 << 32) | sorted_position  (ties -> smaller j first)
  // ---------------------------------------------------------------
#pragma unroll
  for (int q = 0; q < 4; ++q) {
    unsigned j = tid + q * THREADS;
    unsigned r = hash32(row * 0x9E3779B9u + j * 0x85EBCA6Bu + 0x42424242u);
    s_key[j] = (((unsigned long long)r) << 32) | (unsigned long long)j;
  }

#pragma unroll
  for (unsigned size = 2; size <= SEG; size <<= 1) {
#pragma unroll
    for (unsigned stride = size >> 1; stride > 0; stride >>= 1) {
      __syncthreads();
#pragma unroll
      for (int q = 0; q < 2; ++q) {
        unsigned p   = tid + q * THREADS;                      // 0..511
        unsigned seg = p >> 7;                                 // 128 pairs/seg
        unsigned lp  = p & 127u;
        unsigned il  = ((lp & ~(stride - 1u)) << 1) | (lp & (stride - 1u));
        unsigned i   = (seg << 8) + il;
        unsigned jj  = i + stride;
        bool up = ((il & size) == 0u);
        unsigned long long a = s_key[i], b = s_key[jj];
        bool sw = up ? (a > b) : (b > a);
        if (sw) { s_key[i] = b; s_key[jj] = a; }
      }
    }
  }
  __syncthreads();

  // ---------------------------------------------------------------
  // Phase 3: thresholds k[seg] from mask_ratio (replicates the Python
  // double-precision arithmetic). Segment 0 (best scores) uses
  // layer_ratios[3], seg 1 -> [2], etc. Scatter the mask through the
  // permutation INTO LDS (cheap random 4B writes), not into HBM.
  // ---------------------------------------------------------------
  const double mr = (double)p_mask_ratio[0];
  const double strength = 0.15;
  double base = mr - 3.0 * strength * 0.5;
  double ratios[4];
  double wavg = 0.0;
#pragma unroll
  for (int i2 = 0; i2 < 4; ++i2) {
    double rr = base + (double)(3 - i2) * strength;
    rr = rr < 0.0 ? 0.0 : rr;
    rr = rr > 0.9 ? 0.9 : rr;
    ratios[i2] = rr;
    wavg += rr * 0.25;
  }
  if (wavg > 0.0) {
    double sc = mr / wavg;
#pragma unroll
    for (int i2 = 0; i2 < 4; ++i2) ratios[i2] *= sc;
  }
  int kseg[4];
#pragma unroll
  for (int s2 = 0; s2 < 4; ++s2) kseg[s2] = (int)(256.0 * ratios[3 - s2]);

#pragma unroll
  for (int q = 0; q < 4; ++q) {
    unsigned slot = tid + q * THREADS;           // seg*256 + rank
    unsigned seg  = slot >> 8;
    int      rnk  = (int)(slot & 255u);
    unsigned jg   = (unsigned)(s_key[slot] & 0xFFFFFFFFull); // sorted position
    unsigned orig = s_order[jg];                 // original index in row
    s_scores[orig] = (rnk < kseg[seg]) ? 1.0f : 0.0f;        // LDS scatter
  }
  __syncthreads();

  // ---------------------------------------------------------------
  // Phase 4: drain the completed row LDS -> HBM with fully coalesced
  // 16 B/lane async stores (ASYNCcnt). S_ENDPGM performs an implicit
  // wait-idle, but wait explicitly for clean counter state.
  // ---------------------------------------------------------------
  {
    unsigned ldsOff  = (unsigned)(uintptr_t)(&s_scores[0]) + tid * 16u;
    unsigned byteOff = tid * 16u;
    unsigned long long obase =
        (unsigned long long)(uintptr_t)(out + (size_t)row * NROW);
    asm volatile("global_store_async_from_lds_b128 %0, %1, %2"
                 :: "v"(byteOff), "v"(ldsOff), "s"(obase)
                 : "memory");
    asm volatile("s_wait_asynccnt 0" ::: "memory");
  }
}

extern "C" void kernel_launch(void* const* d_in, const int* in_sizes, int n_in,
                              void* d_out, int out_size, void* d_ws, size_t ws_size,
                              hipStream_t stream) {
  (void)n_in; (void)d_ws; (void)ws_size; (void)out_size;
  const float* scores = (const float*)d_in[0];
  const float* mratio = (const float*)d_in[1];
  float* out = (float*)d_out;
  int rows = in_sizes[0] / NROW;   // B*T = 8192
  stratified_mask_kernel<<<rows, THREADS, 0, stream>>>(scores, mratio, out);
}